// MeanPool2_35304631174309
// MI455X (gfx1250) — compile-verified
//
#include <hip/hip_runtime.h>

typedef __attribute__((ext_vector_type(2))) float v2f;
typedef __attribute__((ext_vector_type(8))) float v8f;

#define D_NODE 32
#define D_EDGE 16
#define D_IN   48   // D_NODE + D_EDGE
#define D_OUT  32
#define W_LDW  50   // padded LDS leading dim: conflict-free banks, keeps v2f 8B-aligned

// ---------------------------------------------------------------------------
// Kernel 1: init workspace. acc[N,32] = 0, deg[N] = 1 (self message counts).
// ---------------------------------------------------------------------------
__global__ void gnn_init(float* __restrict__ acc, float* __restrict__ deg,
                         int n_nodes) {
  const int i = blockIdx.x * blockDim.x + threadIdx.x;
  if (i < n_nodes * D_OUT) acc[i] = 0.0f;
  if (i < n_nodes)         deg[i] = 1.0f;
}

// ---------------------------------------------------------------------------
// Kernel 2: per-edge projection + scatter, grid-stride over 16-edge tiles.
// W is transposed into LDS once per block, B fragments hoisted to registers
// once per wave (48 VGPRs), then each tile is: 12 v2f A gathers ->
// 24 v_wmma_f32_16x16x4_f32 -> 32 f32 atomics/edge scatter (vs 48 pre-GEMM).
// ---------------------------------------------------------------------------
__global__ void __launch_bounds__(256)
gnn_edge(const float* __restrict__ h, const float* __restrict__ e,
         const float* __restrict__ W, const int* __restrict__ src,
         const int* __restrict__ dst, float* __restrict__ acc,
         float* __restrict__ deg, int n_edges, int n_tiles) {
  __shared__ float Wt[D_OUT * W_LDW];              // W transposed: Wt[col][k]
  for (int i = threadIdx.x; i < D_IN * D_OUT; i += blockDim.x) {
    const int row = i / D_OUT, col = i % D_OUT;    // coalesced read of W
    Wt[col * W_LDW + row] = W[i];
  }
  __syncthreads();

  const int lane = threadIdx.x & 31;
  const int m    = lane & 15;                      // tile row (M)
  const int half = lane >> 4;                      // K-pair select
  const int koff = half * 2;

  // Hoist all B fragments: one ds_load_b64 each, reused across every tile.
  v2f B0[12], B1[12];
#pragma unroll
  for (int kk = 0; kk < 12; ++kk) {
    const int k = kk * 4 + koff;
    B0[kk] = *(const v2f*)&Wt[m * W_LDW + k];
    B1[kk] = *(const v2f*)&Wt[(16 + m) * W_LDW + k];
  }

  const int waveId = (int)((blockIdx.x * blockDim.x + threadIdx.x) >> 5);
  const int nWaves = (int)((gridDim.x * blockDim.x) >> 5);

  for (int tile = waveId; tile < n_tiles; tile += nWaves) {  // wave-uniform
    const int nt = tile + nWaves;                  // prefetch next stride
    if (nt < n_tiles) {
      __builtin_prefetch(src + nt * 16, 0, 0);
      __builtin_prefetch(dst + nt * 16, 0, 0);
      __builtin_prefetch(e + (long long)nt * 16 * D_EDGE + lane * 8, 0, 0);
    }

    const int tbase   = tile * 16;
    const int rowEdge = tbase + m;
    const int eIdx    = rowEdge < n_edges ? rowEdge : (n_edges - 1);
    const int s       = src[eIdx];
    const int dm      = dst[eIdx];                 // dst of edge row m
    const float* __restrict__ arow = h + (long long)s    * D_NODE;
    const float* __restrict__ erow = e + (long long)eIdx * D_EDGE;

    v2f A[12];
#pragma unroll
    for (int kk = 0; kk < 8; ++kk)                 // node features 0..31
      A[kk] = *(const v2f*)(arow + kk * 4 + koff);
#pragma unroll
    for (int kk = 8; kk < 12; ++kk)                // edge features 32..47
      A[kk] = *(const v2f*)(erow + kk * 4 + koff - D_NODE);

    v8f a0 = {}, a1 = {};
#pragma unroll
    for (int kk = 0; kk < 12; ++kk) {
      a0 = __builtin_amdgcn_wmma_f32_16x16x4_f32(false, A[kk], false, B0[kk],
                                                 (short)0, a0, false, false);
      a1 = __builtin_amdgcn_wmma_f32_16x16x4_f32(false, A[kk], false, B1[kk],
                                                 (short)0, a1, false, false);
    }

    // Scatter: D layout -> VGPR r holds (row = r + 8*half, col = m / m+16).
#pragma unroll
    for (int r = 0; r < 8; ++r) {
      const int row = r + 8 * half;
      const int d   = __shfl(dm, row, 32);         // lane 'row' holds its dst
      if (tbase + row < n_edges) {
        float* p = acc + (long long)d * D_OUT;
        atomicAdd(p + m,      a0[r]);
        atomicAdd(p + 16 + m, a1[r]);
      }
    }
    if (lane < 16 && rowEdge < n_edges)
      atomicAdd(deg + dm, 1.0f);
  }
}

// ---------------------------------------------------------------------------
// Kernel 3: self-message GEMM + mean + bias + L2 normalize, grid-stride.
// Edge accumulator preloaded as WMMA C; h @ W[:32,:] accumulated on top.
// ---------------------------------------------------------------------------
__global__ void __launch_bounds__(256)
gnn_node(const float* __restrict__ h, const float* __restrict__ W,
         const float* __restrict__ bias, const float* __restrict__ accbuf,
         const float* __restrict__ deg, float* __restrict__ out,
         int n_nodes, int n_tiles) {
  __shared__ float Wt[D_OUT * W_LDW];              // top 32 rows, transposed
  for (int i = threadIdx.x; i < D_NODE * D_OUT; i += blockDim.x) {
    const int row = i / D_OUT, col = i % D_OUT;
    Wt[col * W_LDW + row] = W[i];
  }
  __syncthreads();

  const int lane = threadIdx.x & 31;
  const int m    = lane & 15;
  const int half = lane >> 4;
  const int koff = half * 2;

  v2f B0[8], B1[8];
#pragma unroll
  for (int kk = 0; kk < 8; ++kk) {
    const int k = kk * 4 + koff;
    B0[kk] = *(const v2f*)&Wt[m * W_LDW + k];
    B1[kk] = *(const v2f*)&Wt[(16 + m) * W_LDW + k];
  }
  const float bl = bias[m];
  const float bh = bias[16 + m];

  const int waveId = (int)((blockIdx.x * blockDim.x + threadIdx.x) >> 5);
  const int nWaves = (int)((gridDim.x * blockDim.x) >> 5);

  for (int tile = waveId; tile < n_tiles; tile += nWaves) {  // wave-uniform
    const int nbase = tile * 16;

    // Preload accumulated edge contributions as C (same layout as D).
    v8f c0, c1;
#pragma unroll
    for (int r = 0; r < 8; ++r) {
      int nr = nbase + r + 8 * half;
      nr = nr < n_nodes ? nr : (n_nodes - 1);
      c0[r] = accbuf[(long long)nr * D_OUT + m];
      c1[r] = accbuf[(long long)nr * D_OUT + 16 + m];
    }

    const int nA  = (nbase + m) < n_nodes ? (nbase + m) : (n_nodes - 1);
    const float dgm = deg[nA];                     // deg of node row m
    const float* __restrict__ arow = h + (long long)nA * D_NODE;

#pragma unroll
    for (int kk = 0; kk < 8; ++kk) {
      const v2f a = *(const v2f*)(arow + kk * 4 + koff);
      c0 = __builtin_amdgcn_wmma_f32_16x16x4_f32(false, a, false, B0[kk],
                                                 (short)0, c0, false, false);
      c1 = __builtin_amdgcn_wmma_f32_16x16x4_f32(false, a, false, B1[kk],
                                                 (short)0, c1, false, false);
    }

#pragma unroll
    for (int r = 0; r < 8; ++r) {
      const int nr = nbase + r + 8 * half;
      const float rd = 1.0f / __shfl(dgm, r + 8 * half, 32);
      const float v0 = c0[r] * rd + bl;
      const float v1 = c1[r] * rd + bh;
      float ss = v0 * v0 + v1 * v1;
      ss += __shfl_xor(ss, 1, 32);                 // masks <= 8 stay in-half:
      ss += __shfl_xor(ss, 2, 32);                 // reduces the 16 lanes that
      ss += __shfl_xor(ss, 4, 32);                 // together hold one row's
      ss += __shfl_xor(ss, 8, 32);                 // 32 columns
      const float rs = rsqrtf(fmaxf(ss, 1e-12f));
      if (nr < n_nodes) {
        out[(long long)nr * D_OUT + m]      = v0 * rs;
        out[(long long)nr * D_OUT + 16 + m] = v1 * rs;
      }
    }
  }
}

// ---------------------------------------------------------------------------
extern "C" void kernel_launch(void* const* d_in, const int* in_sizes, int n_in,
                              void* d_out, int out_size, void* d_ws, size_t ws_size,
                              hipStream_t stream) {
  const float* h   = (const float*)d_in[0];
  const float* e   = (const float*)d_in[1];
  const float* W   = (const float*)d_in[2];
  const float* b   = (const float*)d_in[3];
  const int*   src = (const int*)d_in[4];
  const int*   dst = (const int*)d_in[5];
  const int n_nodes = in_sizes[0] / D_NODE;
  const int n_edges = in_sizes[4];

  float* acc = (float*)d_ws;                       // [N, 32] edge-GEMM sums
  float* deg = acc + (size_t)n_nodes * D_OUT;      // [N] degree (+1 self)

  {
    const int total = n_nodes * D_OUT;
    gnn_init<<<(total + 255) / 256, 256, 0, stream>>>(acc, deg, n_nodes);
  }
  {
    const int nTiles = (n_edges + 15) / 16;        // 100k tiles
    const int wpb = 8;                             // 256 threads = 8 waves
    int blocks = (nTiles + wpb - 1) / wpb;
    if (blocks > 2048) blocks = 2048;              // ~6 tiles per wave
    gnn_edge<<<blocks, wpb * 32, 0, stream>>>(h, e, W, src, dst, acc, deg,
                                              n_edges, nTiles);
  }
  {
    const int nTiles = (n_nodes + 15) / 16;        // 6250 tiles
    const int wpb = 8;
    int blocks = (nTiles + wpb - 1) / wpb;
    if (blocks > 512) blocks = 512;                // ~1.5 tiles per wave
    gnn_node<<<blocks, wpb * 32, 0, stream>>>(h, W, b, acc, deg,
                                              (float*)d_out, n_nodes, nTiles);
  }
}